// MyModel_61933428415847
// MI455X (gfx1250) — compile-verified
//
#include <hip/hip_runtime.h>

// ---------------------------------------------------------------------------
// Fused LSTM (I=10, H=20, B=4096, T=512) for gfx1250 (MI455X), f32 end-to-end.
//
// Per wave: one 16-row batch tile. Per timestep:
//   gates[16x80] = [x_t(16x10) | 0 | h(16x20)] (K=32)  x  Wc[32x80]  + bias
// computed with V_WMMA_F32_16X16X4_F32 (5 N-tiles x 8 K-chunks), then
// sigmoid/tanh cell update using gfx1250's native V_TANH_F32, with c kept in
// registers and h staged in wave-private LDS.
// ---------------------------------------------------------------------------

typedef float v2f __attribute__((ext_vector_type(2)));
typedef float v8f __attribute__((ext_vector_type(8)));

#define T_STEPS 512
#define IN_F    10
#define HID     20
#define GATES   80
#define ROWS_PER_WAVE   16
#define WAVES_PER_BLOCK 4
#define THREADS_PER_BLOCK (WAVES_PER_BLOCK * 32)

// gfx1250 native transcendental tanh (1 trans op instead of exp + divide).
__device__ __forceinline__ float tanh_hw(float x) {
#if __has_builtin(__builtin_amdgcn_tanhf)
    return __builtin_amdgcn_tanhf(x);
#else
    float r;
    asm("v_tanh_f32 %0, %1" : "=v"(r) : "v"(x));
    return r;
#endif
}

// sigmoid(x) = 0.5 * tanh(x/2) + 0.5  -> v_tanh_f32 + 2 FMA-class VALU ops.
__device__ __forceinline__ float sigmoid_hw(float x) {
    return __builtin_fmaf(0.5f, tanh_hw(0.5f * x), 0.5f);
}

__global__ void __launch_bounds__(THREADS_PER_BLOCK, 1)
lstm_fused_kernel(const float* __restrict__ data,
                  const float* __restrict__ W_ih,
                  const float* __restrict__ W_hh,
                  const float* __restrict__ b_ih,
                  const float* __restrict__ b_hh,
                  float* __restrict__ block_partials)
{
    // Wave-private LDS tiles (no cross-wave sharing; per-wave LDS is in-order).
    __shared__ __align__(16) float gls[WAVES_PER_BLOCK][16][GATES]; // gates 16x80
    __shared__ __align__(16) float hls[WAVES_PER_BLOCK][16][HID];   // h     16x20
    __shared__ float red[THREADS_PER_BLOCK];

    const int  tid  = threadIdx.x;
    const int  lane = tid & 31;
    const int  wave = tid >> 5;
    const int  ln   = lane & 15;          // lane within half-wave
    const bool hi   = (lane >= 16);       // upper half-wave
    const int  rowBase = blockIdx.x * (WAVES_PER_BLOCK * ROWS_PER_WAVE)
                       + wave * ROWS_PER_WAVE;

    // Cell ownership: each lane owns a contiguous 10-wide run of hidden units
    // in one batch row (same global mapping p = lane*10+q, expressed so the
    // LDS gate gather / h scatter vectorize to b64 accesses).
    const int m  = lane >> 1;             // batch row within tile (0..15)
    const int j0 = (lane & 1) * 10;       // first hidden unit of this lane's run

    // --- init: c = 0 in registers, h = 0 in LDS ---
    float c[10];
#pragma unroll
    for (int q = 0; q < 10; q += 2) {
        c[q] = 0.0f; c[q + 1] = 0.0f;
        *reinterpret_cast<float2*>(&hls[wave][m][j0 + q]) = float2{0.0f, 0.0f};
    }

    // --- build B fragments once: Wc[32][80], rows 0..9 = W_ih^T,
    //     rows 10..11 = 0 (K padding), rows 12..31 = W_hh^T.
    v2f   bfrag[5][8];
    float biasv[5];
#pragma unroll
    for (int n = 0; n < 5; ++n) {
        const int col = n * 16 + ln;                     // gate column 0..79
        biasv[n] = b_ih[col] + b_hh[col];
#pragma unroll
        for (int k = 0; k < 8; ++k) {
            const int r0 = 4 * k + (hi ? 2 : 0);
            const int r1 = r0 + 1;
            float w0 = (r0 < IN_F) ? W_ih[col * IN_F + r0]
                     : (r0 < 12)   ? 0.0f
                     :               W_hh[col * HID + (r0 - 12)];
            float w1 = (r1 < IN_F) ? W_ih[col * IN_F + r1]
                     : (r1 < 12)   ? 0.0f
                     :               W_hh[col * HID + (r1 - 12)];
            bfrag[n][k].x = w0;
            bfrag[n][k].y = w1;
        }
    }

    // Per-lane streaming pointer into data[row, :, :] (row-major [B,T,I]).
    const float* xrow = data + (size_t)(rowBase + ln) * T_STEPS * IN_F;
    const int xo0 = hi ? 2 : 0;   // A-chunk 0 column pair
    const int xo1 = hi ? 6 : 4;   // A-chunk 1 column pair
    const int ho  = hi ? 2 : 0;   // h column pair base (chunks 3..7)

    const v2f zero2 = {0.0f, 0.0f};
    float lsum = 0.0f;

    for (int t = 0; t < T_STEPS; ++t) {
        // ---- A operand: x_t part (aligned b64 loads from HBM) ----
        const float* xt = xrow + t * IN_F;
        float2 x0 = *reinterpret_cast<const float2*>(xt + xo0);
        float2 x1 = *reinterpret_cast<const float2*>(xt + xo1);
        float2 x2 = *reinterpret_cast<const float2*>(xt + 8); // cols 8,9

        v2f a[8];
        a[0] = v2f{x0.x, x0.y};
        a[1] = v2f{x1.x, x1.y};
        v2f x2v = {x2.x, x2.y};
        a[2] = hi ? zero2 : x2v;  // upper half would be pad cols 10,11 -> 0

        // ---- A operand: h part from LDS (b64 reads, row stride 20 f32) ----
        const float* hp = &hls[wave][ln][0];
#pragma unroll
        for (int k = 0; k < 5; ++k) {
            float2 hv = *reinterpret_cast<const float2*>(hp + 4 * k + ho);
            a[3 + k] = v2f{hv.x, hv.y};
        }

        // ---- gates = A x Wc + bias, five 16x16 f32 tiles ----
#pragma unroll
        for (int n = 0; n < 5; ++n) {
            const float bv = biasv[n];
            v8f acc = {bv, bv, bv, bv, bv, bv, bv, bv};
#pragma unroll
            for (int k = 0; k < 8; ++k) {
                acc = __builtin_amdgcn_wmma_f32_16x16x4_f32(
                        false, a[k], false, bfrag[n][k],
                        (short)0, acc, false, false);
            }
            // C/D layout: VGPR v -> row M = v (+8 for upper half), N = ln.
#pragma unroll
            for (int v = 0; v < 8; ++v)
                gls[wave][v + (hi ? 8 : 0)][n * 16 + ln] = acc[v];
        }

        // Same-wave LDS is processed in-order; the wait + clobber pins the
        // compiler from hoisting the gather loads above the stores.
        asm volatile("s_wait_dscnt 0" ::: "memory");

        // ---- per-cell gate nonlinearities + state update (10 cells/lane) ----
        const float* gp = &gls[wave][m][j0];
        float hnew[10];
#pragma unroll
        for (int q = 0; q < 10; q += 2) {
            float2 iv2 = *reinterpret_cast<const float2*>(gp + q);
            float2 fv2 = *reinterpret_cast<const float2*>(gp + 20 + q);
            float2 gv2 = *reinterpret_cast<const float2*>(gp + 40 + q);
            float2 ov2 = *reinterpret_cast<const float2*>(gp + 60 + q);

            float i0 = sigmoid_hw(iv2.x), i1 = sigmoid_hw(iv2.y);
            float f0 = sigmoid_hw(fv2.x), f1 = sigmoid_hw(fv2.y);
            float g0 = tanh_hw(gv2.x),    g1 = tanh_hw(gv2.y);
            float o0 = sigmoid_hw(ov2.x), o1 = sigmoid_hw(ov2.y);

            c[q]     = __builtin_fmaf(f0, c[q],     i0 * g0);
            c[q + 1] = __builtin_fmaf(f1, c[q + 1], i1 * g1);
            float h0 = o0 * tanh_hw(c[q]);
            float h1 = o1 * tanh_hw(c[q + 1]);
            lsum += h0 + h1;
            hnew[q] = h0; hnew[q + 1] = h1;
        }
        float* hpw = &hls[wave][m][j0];
#pragma unroll
        for (int q = 0; q < 10; q += 2)
            *reinterpret_cast<float2*>(hpw + q) = float2{hnew[q], hnew[q + 1]};

        asm volatile("s_wait_dscnt 0" ::: "memory");
    }

    // ---- deterministic block reduction ----
    red[tid] = lsum;
    __syncthreads();
#pragma unroll
    for (int s = THREADS_PER_BLOCK / 2; s > 0; s >>= 1) {
        if (tid < s) red[tid] += red[tid + s];
        __syncthreads();
    }
    if (tid == 0) block_partials[blockIdx.x] = red[0];
}

__global__ void final_reduce_kernel(const float* __restrict__ partials,
                                    float* __restrict__ out, int n)
{
    if (threadIdx.x == 0 && blockIdx.x == 0) {
        float s = 0.0f;
        for (int i = 0; i < n; ++i) s += partials[i];   // fixed order
        out[0] = s;
    }
}

extern "C" void kernel_launch(void* const* d_in, const int* in_sizes, int n_in,
                              void* d_out, int out_size, void* d_ws, size_t ws_size,
                              hipStream_t stream)
{
    (void)in_sizes; (void)n_in; (void)out_size; (void)ws_size;
    const float* data = (const float*)d_in[0];
    const float* W_ih = (const float*)d_in[1];
    const float* W_hh = (const float*)d_in[2];
    const float* b_ih = (const float*)d_in[3];
    const float* b_hh = (const float*)d_in[4];
    float* out      = (float*)d_out;
    float* partials = (float*)d_ws;

    const int B = 4096;
    const int blocks = B / (WAVES_PER_BLOCK * ROWS_PER_WAVE); // 64

    lstm_fused_kernel<<<blocks, THREADS_PER_BLOCK, 0, stream>>>(
        data, W_ih, W_hh, b_ih, b_hh, partials);
    final_reduce_kernel<<<1, 32, 0, stream>>>(partials, out, blocks);
}